// GSAGE_Net_65163243815283
// MI455X (gfx1250) — compile-verified
//
#include <hip/hip_runtime.h>

#define NNODES 50000
#define FDIM   256
#define CDIM   40

typedef __attribute__((ext_vector_type(2))) float v2f;
typedef __attribute__((ext_vector_type(8))) float v8f;

__global__ void zero_kernel(float* __restrict__ p, long long n) {
    long long i = (long long)blockIdx.x * blockDim.x + threadIdx.x;
    if (i < n) p[i] = 0.0f;
}

__global__ void deg_kernel(const int* __restrict__ dst, float* __restrict__ deg, int E) {
    int e = blockIdx.x * blockDim.x + threadIdx.x;
    if (e < E) atomicAdd(&deg[dst[e]], 1.0f);
}

__global__ void invdeg_kernel(const float* __restrict__ deg, float* __restrict__ invdeg, int n) {
    int i = blockIdx.x * blockDim.x + threadIdx.x;
    if (i < n) invdeg[i] = 1.0f / fmaxf(deg[i], 1.0f);
}

// Bt[NPAD][K] = transpose of B[K][N], zero-padded in n. Makes the per-lane
// WMMA B-fragment (B[k][col], B[k+1][col]) a contiguous 64-bit load and
// removes all bounds checks from the GEMM inner loop.
__global__ void transpose_pad_kernel(const float* __restrict__ B, float* __restrict__ Bt,
                                     int K, int N, int NPAD) {
    int idx = blockIdx.x * blockDim.x + threadIdx.x;
    if (idx >= NPAD * K) return;
    int n = idx / K;
    int k = idx - n * K;
    Bt[idx] = (n < N) ? B[(long long)k * N + n] : 0.0f;
}

// One thread handles 4 consecutive features of one edge: float4 gather from
// feat[src], 4 atomic f32 adds into agg[dst]. A wave covers 128 contiguous
// floats of one edge -> coalesced; node features live in the 192MB L2.
__global__ void scatter_kernel(const float* __restrict__ feat,
                               const int* __restrict__ src,
                               const int* __restrict__ dst,
                               float* __restrict__ agg, int E) {
    long long tid = (long long)blockIdx.x * blockDim.x + threadIdx.x;
    if (tid >= (long long)E * (FDIM / 4)) return;
    int e = (int)(tid >> 6);             // FDIM/4 == 64 chunks per edge
    int c = ((int)tid & 63) << 2;
    int s = src[e], d = dst[e];
    float4 v = *(const float4*)(feat + (long long)s * FDIM + c);
    float* out = agg + (long long)d * FDIM + c;
    atomicAdd(out + 0, v.x);
    atomicAdd(out + 1, v.y);
    atomicAdd(out + 2, v.z);
    atomicAdd(out + 3, v.w);
}

// C[M x NOUT] = elu?( (invdeg[m]*A1[m,:]) @ B1 + A2 @ B2 + bias ), fp32 WMMA.
// B matrices are pre-transposed+padded (Bt[NPAD][K]). One wave32 per 16x16
// output tile; branch-free inner loop: 4x global_load_b64 + v_pk_mul_f32 +
// 2x v_wmma_f32_16x16x4_f32. EXEC stays all-ones through every WMMA.
template<int NPAD, int NOUT, bool ELU, int K>
__global__ void __launch_bounds__(256)
gemm_dual_wmma(const float* __restrict__ A1, const float* __restrict__ Bt1,
               const float* __restrict__ A2, const float* __restrict__ Bt2,
               const float* __restrict__ bias, const float* __restrict__ invdeg,
               float* __restrict__ C, int M) {
    constexpr int NTN = NPAD / 16;
    int gwave = (int)(((long long)blockIdx.x * blockDim.x + threadIdx.x) >> 5);
    int lane  = threadIdx.x & 31;
    int ntm = M >> 4;                       // M is a multiple of 16 (50000)
    if (gwave >= ntm * NTN) return;         // wave-uniform: EXEC stays all-ones
    int tm = gwave / NTN;
    int tn = gwave - tm * NTN;

    int half = lane >> 4;                   // selects K pair {k,k+1} vs {k+2,k+3}
    int l16  = lane & 15;
    int arow = tm * 16 + l16;               // A row this lane feeds
    int col  = tn * 16 + l16;               // B/C column this lane feeds (< NPAD)

    float scale = invdeg[arow];             // mean-aggregation normalization
    const float* a1p = A1  + (long long)arow * K;
    const float* a2p = A2  + (long long)arow * K;
    const float* b1p = Bt1 + (long long)col  * K;
    const float* b2p = Bt2 + (long long)col  * K;

    v8f acc = {};
#pragma unroll 8
    for (int k = 0; k < K; k += 4) {
        int ka = k + half * 2;
        v2f a1 = *(const v2f*)(a1p + ka);
        a1 = a1 * scale;
        v2f b1 = *(const v2f*)(b1p + ka);
        acc = __builtin_amdgcn_wmma_f32_16x16x4_f32(false, a1, false, b1,
                                                    (short)0, acc, false, false);
        v2f a2 = *(const v2f*)(a2p + ka);
        v2f b2 = *(const v2f*)(b2p + ka);
        acc = __builtin_amdgcn_wmma_f32_16x16x4_f32(false, a2, false, b2,
                                                    (short)0, acc, false, false);
    }

    bool colok = (NPAD == NOUT) ? true : (col < NOUT);
    float bv = colok ? bias[col] : 0.0f;
#pragma unroll
    for (int r = 0; r < 8; ++r) {
        int orow = tm * 16 + half * 8 + r;  // C/D VGPR layout (ISA 7.12.2)
        float v = acc[r] + bv;
        if (ELU) v = (v > 0.0f) ? v : expm1f(v);
        if (colok) C[(long long)orow * NOUT + col] = v;
    }
}

extern "C" void kernel_launch(void* const* d_in, const int* in_sizes, int n_in,
                              void* d_out, int out_size, void* d_ws, size_t ws_size,
                              hipStream_t stream) {
    const float* x   = (const float*)d_in[0];
    const int*   ei  = (const int*)d_in[1];
    const float* W1l = (const float*)d_in[2];
    const float* b1  = (const float*)d_in[3];
    const float* W1r = (const float*)d_in[4];
    const float* W2l = (const float*)d_in[5];
    const float* b2  = (const float*)d_in[6];
    const float* W2r = (const float*)d_in[7];
    const float* Q   = (const float*)d_in[8];
    float* out = (float*)d_out;

    const int E = in_sizes[1] / 2;          // edge_index is [2, E] int32
    const int* src = ei;
    const int* dst = ei + E;

    const int C2PAD = 48;                   // CDIM=40 padded to 3 col tiles

    // Workspace (floats): agg | h | deg | invdeg | Bt1l | Bt1r | Bt2l | Bt2r
    float* agg    = (float*)d_ws;
    float* h      = agg    + (long long)NNODES * FDIM;
    float* deg    = h      + (long long)NNODES * FDIM;
    float* invdeg = deg    + NNODES;
    float* Bt1l   = invdeg + NNODES;
    float* Bt1r   = Bt1l   + FDIM * FDIM;
    float* Bt2l   = Bt1r   + FDIM * FDIM;
    float* Bt2r   = Bt2l   + C2PAD * FDIM;

    const int BT = 256;
    const long long aggN = (long long)NNODES * FDIM;

    // ---- weight transpose/pad (removes all guards from GEMM K-loops) ----
    transpose_pad_kernel<<<(FDIM * FDIM + BT - 1) / BT, BT, 0, stream>>>(W1l, Bt1l, FDIM, FDIM, FDIM);
    transpose_pad_kernel<<<(FDIM * FDIM + BT - 1) / BT, BT, 0, stream>>>(W1r, Bt1r, FDIM, FDIM, FDIM);
    transpose_pad_kernel<<<(C2PAD * FDIM + BT - 1) / BT, BT, 0, stream>>>(W2l, Bt2l, FDIM, CDIM, C2PAD);
    transpose_pad_kernel<<<(C2PAD * FDIM + BT - 1) / BT, BT, 0, stream>>>(W2r, Bt2r, FDIM, CDIM, C2PAD);

    // ---- degree (shared by both layers) ----
    zero_kernel<<<(NNODES + BT - 1) / BT, BT, 0, stream>>>(deg, NNODES);
    deg_kernel<<<(E + BT - 1) / BT, BT, 0, stream>>>(dst, deg, E);
    invdeg_kernel<<<(NNODES + BT - 1) / BT, BT, 0, stream>>>(deg, invdeg, NNODES);

    // ---- layer 1: agg = segsum(x[src] -> dst); h = elu(agg/deg @ W1l + x @ W1r + b1) ----
    zero_kernel<<<(int)((aggN + BT - 1) / BT), BT, 0, stream>>>(agg, aggN);
    {
        long long t = (long long)E * (FDIM / 4);
        scatter_kernel<<<(int)((t + BT - 1) / BT), BT, 0, stream>>>(x, src, dst, agg, E);
    }
    {
        int waves = (NNODES / 16) * (FDIM / 16);
        int blocks = (int)(((long long)waves * 32 + BT - 1) / BT);
        gemm_dual_wmma<FDIM, FDIM, true, FDIM><<<blocks, BT, 0, stream>>>(
            agg, Bt1l, x, Bt1r, b1, invdeg, h, NNODES);
    }

    // ---- layer 2: agg = segsum(h[src] -> dst); out = agg/deg @ W2l + h @ W2r + b2 ----
    zero_kernel<<<(int)((aggN + BT - 1) / BT), BT, 0, stream>>>(agg, aggN);
    {
        long long t = (long long)E * (FDIM / 4);
        scatter_kernel<<<(int)((t + BT - 1) / BT), BT, 0, stream>>>(h, src, dst, agg, E);
    }
    {
        int waves = (NNODES / 16) * (C2PAD / 16);
        int blocks = (int)(((long long)waves * 32 + BT - 1) / BT);
        gemm_dual_wmma<C2PAD, CDIM, false, FDIM><<<blocks, BT, 0, stream>>>(
            agg, Bt2l, h, Bt2r, b2, invdeg, out, NNODES);
    }

    // ---- pass-through Q into second half of the tuple output ----
    hipMemcpyAsync(out + (long long)NNODES * CDIM, Q,
                   (size_t)in_sizes[8] * sizeof(float),
                   hipMemcpyDeviceToDevice, stream);
}